// DeformConvAddReLU62d_15719580304169
// MI455X (gfx1250) — compile-verified
//
#include <hip/hip_runtime.h>
#include <hip/hip_bf16.h>

typedef __attribute__((ext_vector_type(2))) float v2f;
typedef __attribute__((ext_vector_type(8))) float v8f;

// Problem constants
#define BB    8
#define CIN   64
#define COUT  64
#define HH    96
#define WW    96
#define KK9   9
#define KDIM  576            // CIN * 9
#define TILE_N 16            // pixels per tile (one row segment)
#define PLANE (HH*WW)        // 9216

__global__ __launch_bounds__(128)
void deform_conv_wmma_kernel(const float* __restrict__ x,
                             const float* __restrict__ offs,
                             const float* __restrict__ x2,
                             const float* __restrict__ weight,
                             const float* __restrict__ bias,
                             float* __restrict__ out)
{
    // B-matrix staged as K-pairs: lB2[kpair*16 + pix] = {K=2*kpair, K=2*kpair+1}
    __shared__ v2f   lB2[(KDIM/2) * TILE_N];   // 288*16*8B = 36 KB
    __shared__ int   lY0[KK9 * TILE_N];
    __shared__ int   lX0[KK9 * TILE_N];
    __shared__ float lFY[KK9 * TILE_N];
    __shared__ float lFX[KK9 * TILE_N];

    const int tid = threadIdx.x;
    const int blk = blockIdx.x;                 // 0 .. 8*96*6-1
    const int b   = blk / (HH * (WW / TILE_N));
    const int rem = blk % (HH * (WW / TILE_N));
    const int ho  = rem / (WW / TILE_N);
    const int wo0 = (rem % (WW / TILE_N)) * TILE_N;

    // ---------- Phase 1: sampling coordinates for 144 (k, pix) items ----------
    for (int i = tid; i < KK9 * TILE_N; i += 128) {
        const int pix = i & (TILE_N - 1);
        const int kd  = i >> 4;                              // 0..8
        const int obase = ((b * 18 + kd * 2) * HH + ho) * WW + wo0 + pix;
        const float oy = offs[obase];                        // channel 2k   (y)
        const float ox = offs[obase + PLANE];                // channel 2k+1 (x)
        const float py = (float)(ho  - 1 + kd / 3) + oy;
        const float px = (float)(wo0 + pix - 1 + kd % 3) + ox;
        const float fy0 = floorf(py);
        const float fx0 = floorf(px);
        lY0[i] = (int)fy0;  lX0[i] = (int)fx0;
        lFY[i] = py - fy0;  lFX[i] = px - fx0;
    }
    __syncthreads();

    // ---------- Phase 2: bilinear gather -> LDS B matrix (576 x 16) ----------
    float* lBf = (float*)lB2;
#pragma unroll 2
    for (int i = tid; i < KDIM * TILE_N; i += 128) {
        const int pix = i & (TILE_N - 1);
        const int kt  = i >> 4;                  // K index = cin*9 + kd, 0..575
        const int cin = kt / 9;
        const int kd  = kt - cin * 9;
        const int ci  = kd * TILE_N + pix;       // coord slot

        const int   y0 = lY0[ci];
        const int   x0 = lX0[ci];
        const float fy = lFY[ci];
        const float fx = lFX[ci];
        const float gy = 1.0f - fy, gx = 1.0f - fx;

        const float vy0 = (y0 >= 0  && y0 <  HH) ? 1.0f : 0.0f;
        const float vy1 = (y0 >= -1 && y0 <  HH - 1) ? 1.0f : 0.0f;
        const float vx0 = (x0 >= 0  && x0 <  WW) ? 1.0f : 0.0f;
        const float vx1 = (x0 >= -1 && x0 <  WW - 1) ? 1.0f : 0.0f;

        const int y0c = min(max(y0, 0), HH - 1);
        const int y1c = min(max(y0 + 1, 0), HH - 1);
        const int x0c = min(max(x0, 0), WW - 1);
        const int x1c = min(max(x0 + 1, 0), WW - 1);

        const float* __restrict__ xp = x + (size_t)(b * CIN + cin) * PLANE;
        const float s =
            (gy * gx * vy0 * vx0) * xp[y0c * WW + x0c] +
            (gy * fx * vy0 * vx1) * xp[y0c * WW + x1c] +
            (fy * gx * vy1 * vx0) * xp[y1c * WW + x0c] +
            (fy * fx * vy1 * vx1) * xp[y1c * WW + x1c];

        lBf[(kt >> 1) * (2 * TILE_N) + pix * 2 + (kt & 1)] = s;
    }
    __syncthreads();

    // ---------- Phase 3: WMMA f32 16x16x4 GEMM, one 16-Cout tile per wave ----
    const int lane = tid & 31;
    const int wv   = tid >> 5;                    // wave id 0..3
    const int m    = lane & 15;                   // row-in-tile / pixel lane
    const int hi   = lane >> 4;                   // 0: K+0/1 , 1: K+2/3

    // A fragment source: weight[(coutBase+m)][kk*4 + 2*hi .. +1]
    const float* __restrict__ wptr =
        weight + (size_t)(wv * 16 + m) * KDIM + hi * 2;

    // Two interleaved accumulators halve the D->C dependency chain depth;
    // both carry 16x16 tiles in identical layout, summed exactly at the end.
    v8f acc0 = {0.f, 0.f, 0.f, 0.f, 0.f, 0.f, 0.f, 0.f};
    v8f acc1 = {0.f, 0.f, 0.f, 0.f, 0.f, 0.f, 0.f, 0.f};

#pragma unroll 4
    for (int kk = 0; kk < KDIM / 4; kk += 2) {
        v2f a0 = *(const v2f*)(wptr + kk * 4);
        v2f a1 = *(const v2f*)(wptr + kk * 4 + 4);
        v2f b0 = lB2[(kk * 2 + hi) * TILE_N + m];
        v2f b1 = lB2[(kk * 2 + 2 + hi) * TILE_N + m];
        acc0 = __builtin_amdgcn_wmma_f32_16x16x4_f32(
            false, a0, false, b0, (short)0, acc0, false, false);
        acc1 = __builtin_amdgcn_wmma_f32_16x16x4_f32(
            false, a1, false, b1, (short)0, acc1, false, false);
    }

    // ---------- Epilogue: bias + residual + ReLU6 -----------------------------
    const int wo = wo0 + m;
#pragma unroll
    for (int v = 0; v < 8; ++v) {
        const int cout = wv * 16 + hi * 8 + v;    // D layout: M = v (+8 for hi)
        const int oidx = ((b * COUT + cout) * HH + ho) * WW + wo;
        float r = (acc0[v] + acc1[v]) + bias[cout] + x2[oidx];
        r = fminf(fmaxf(r, 0.0f), 6.0f);
        out[oidx] = r;
    }
}

extern "C" void kernel_launch(void* const* d_in, const int* in_sizes, int n_in,
                              void* d_out, int out_size, void* d_ws, size_t ws_size,
                              hipStream_t stream) {
    const float* x      = (const float*)d_in[0];
    const float* offs   = (const float*)d_in[1];
    const float* x2     = (const float*)d_in[2];
    const float* weight = (const float*)d_in[3];
    const float* bias   = (const float*)d_in[4];
    float* out          = (float*)d_out;

    const int grid = BB * HH * (WW / TILE_N);     // 8*96*6 = 4608
    deform_conv_wmma_kernel<<<grid, 128, 0, stream>>>(x, offs, x2, weight, bias, out);
}